// Entmax15_49615462203786
// MI455X (gfx1250) — compile-verified
//
#include <hip/hip_runtime.h>

#define ROW 1024
#define TPB 256
#define EPT 4   // ROW / TPB elements per thread

typedef int   v4i __attribute__((ext_vector_type(4)));
typedef float v4f __attribute__((ext_vector_type(4)));
typedef __attribute__((address_space(1))) v4i gv4i;   // global
typedef __attribute__((address_space(3))) v4i lv4i;   // LDS

__device__ __forceinline__ void cmpswap(float& a, float& b, bool desc) {
  float lo = fminf(a, b);
  float hi = fmaxf(a, b);
  a = desc ? hi : lo;
  b = desc ? lo : hi;
}

__global__ __launch_bounds__(TPB) void entmax15_fwd_kernel(const float* __restrict__ X,
                                                           float* __restrict__ Y) {
  __shared__ __align__(16) float s_raw[ROW];   // raw input row (never modified)
  __shared__ __align__(16) float s_srt[ROW];   // shifted/halved -> sorted descending
  __shared__ float s_tau[ROW];                 // tau per sorted position
  __shared__ float s_r1[8];                    // per-wave partials (sum x / max)
  __shared__ float s_r2[8];                    // per-wave partials (sum x^2)
  __shared__ int   s_ci[8];                    // per-wave support counts

  const int tid  = threadIdx.x;
  const int lane = tid & 31;
  const int wid  = tid >> 5;                   // 8 wave32 waves
  const int base = tid * EPT;
  const long long row = blockIdx.x;
  const float* gsrc = X + row * (long long)ROW + base;
  float* ldst = &s_raw[base];

  // ---- 1) Stage the 4KB row into LDS via the CDNA5 async copy path ----
#if __has_builtin(__builtin_amdgcn_global_load_async_to_lds_b128)
  __builtin_amdgcn_global_load_async_to_lds_b128(
      (gv4i*)(void*)gsrc, (lv4i*)(void*)ldst, 0, 0);
#if __has_builtin(__builtin_amdgcn_s_wait_asynccnt)
  __builtin_amdgcn_s_wait_asynccnt(0);
#else
  asm volatile("s_wait_asynccnt 0" ::: "memory");
#endif
#else
  *(v4f*)ldst = *(const v4f*)gsrc;
#endif
  __syncthreads();

  // ---- 2) Row max: wave32 xor-shuffle reduce + 8-entry LDS handoff ----
  v4f r = *(const v4f*)&s_raw[base];
  float m = fmaxf(fmaxf(r.x, r.y), fmaxf(r.z, r.w));
#pragma unroll
  for (int off = 16; off > 0; off >>= 1) m = fmaxf(m, __shfl_xor(m, off, 32));
  if (lane == 0) s_r1[wid] = m;
  __syncthreads();
  float rowmax = s_r1[0];
#pragma unroll
  for (int w = 1; w < 8; ++w) rowmax = fmaxf(rowmax, s_r1[w]);

  // ---- 3) Shift + halve in registers; bitonic k=2 and k=4 fully in registers ----
  float x0 = (r.x - rowmax) * 0.5f;
  float x1 = (r.y - rowmax) * 0.5f;
  float x2 = (r.z - rowmax) * 0.5f;
  float x3 = (r.w - rowmax) * 0.5f;

  // k=2, j=1: dir = ((i & 2) == 0)
  cmpswap(x0, x1, true);
  cmpswap(x2, x3, false);
  // k=4: dir constant over the chunk
  {
    bool d4 = ((base & 4) == 0);
    cmpswap(x0, x2, d4); cmpswap(x1, x3, d4);   // j=2
    cmpswap(x0, x1, d4); cmpswap(x2, x3, d4);   // j=1
  }
  s_srt[base + 0] = x0; s_srt[base + 1] = x1;
  s_srt[base + 2] = x2; s_srt[base + 3] = x3;

  // ---- 4) Bitonic k=8..1024: cross-chunk stages in LDS, j<=2 in registers ----
  for (int k = 8; k <= ROW; k <<= 1) {
    for (int j = k >> 1; j >= 4; j >>= 1) {
      __syncthreads();
#pragma unroll
      for (int e = 0; e < EPT; ++e) {
        int i = tid + e * TPB;          // covers 0..ROW-1
        int ixj = i ^ j;
        if (ixj > i) {
          float a = s_srt[i];
          float b = s_srt[ixj];
          bool descBlock = ((i & k) == 0);
          bool doSwap = descBlock ? (a < b) : (a > b);
          if (doSwap) { s_srt[i] = b; s_srt[ixj] = a; }
        }
      }
    }
    __syncthreads();
    x0 = s_srt[base + 0]; x1 = s_srt[base + 1];
    x2 = s_srt[base + 2]; x3 = s_srt[base + 3];
    bool dk = ((base & k) == 0);
    cmpswap(x0, x2, dk); cmpswap(x1, x3, dk);   // j=2
    cmpswap(x0, x1, dk); cmpswap(x2, x3, dk);   // j=1
    s_srt[base + 0] = x0; s_srt[base + 1] = x1;
    s_srt[base + 2] = x2; s_srt[base + 3] = x3;
  }
  // x0..x3 now hold this thread's sorted chunk (descending order, positions base..base+3)

  // ---- 5) Scan of x and x^2: wave32 shfl_up scan + 8-entry LDS handoff ----
  float l1 = x0 + x1 + x2 + x3;
  float l2 = x0 * x0 + x1 * x1 + x2 * x2 + x3 * x3;
  float s1 = l1, s2 = l2;
#pragma unroll
  for (int off = 1; off < 32; off <<= 1) {
    float a1 = __shfl_up(s1, off, 32);
    float a2 = __shfl_up(s2, off, 32);
    if (lane >= off) { s1 += a1; s2 += a2; }
  }
  if (lane == 31) { s_r1[wid] = s1; s_r2[wid] = s2; }
  __syncthreads();
  float woff1 = 0.0f, woff2 = 0.0f;
  for (int w = 0; w < wid; ++w) { woff1 += s_r1[w]; woff2 += s_r2[w]; }
  float run1 = woff1 + s1 - l1;   // exclusive prefix (sum x) at position base
  float run2 = woff2 + s2 - l2;   // exclusive prefix (sum x^2)

  // ---- 6) tau per sorted position + support count ----
  float xs[EPT] = {x0, x1, x2, x3};
  int cnt = 0;
#pragma unroll
  for (int e = 0; e < EPT; ++e) {
    int p = base + e;
    float x = xs[e];
    run1 += x;
    run2 += x * x;
    float rho = (float)(p + 1);
    float mean = run1 / rho;
    float meansq = run2 / rho;
    float ss = rho * (meansq - mean * mean);
    float delta = (1.0f - ss) / rho;
    delta = delta > 0.0f ? delta : 0.0f;
    float tau = mean - sqrtf(delta);
    s_tau[p] = tau;
    cnt += (tau <= x) ? 1 : 0;
  }
#pragma unroll
  for (int off = 16; off > 0; off >>= 1) cnt += __shfl_xor(cnt, off, 32);
  if (lane == 0) s_ci[wid] = cnt;
  __syncthreads();
  int support = 0;
#pragma unroll
  for (int w = 0; w < 8; ++w) support += s_ci[w];
  const float tau_star = s_tau[support - 1];   // support >= 1 always

  // ---- 7) Output: Y = relu(X - tau*)^2 in ORIGINAL order, non-temporal ----
  v4f o;
  float rr[EPT] = {r.x, r.y, r.z, r.w};
#pragma unroll
  for (int e = 0; e < EPT; ++e) {
    float xx = (rr[e] - rowmax) * 0.5f;
    float t = xx - tau_star;
    t = t > 0.0f ? t : 0.0f;
    o[e] = t * t;
  }
  __builtin_nontemporal_store(o, (v4f*)(Y + row * (long long)ROW + base));
}

extern "C" void kernel_launch(void* const* d_in, const int* in_sizes, int n_in,
                              void* d_out, int out_size, void* d_ws, size_t ws_size,
                              hipStream_t stream) {
  const float* X = (const float*)d_in[0];
  float* Y = (float*)d_out;
  const int rows = in_sizes[0] / ROW;   // 8*12*1024 = 98304 rows of 1024
  entmax15_fwd_kernel<<<dim3(rows), dim3(TPB), 0, stream>>>(X, Y);
}